// HeteroGraphClassificationModel_24661702214221
// MI455X (gfx1250) — compile-verified
//
#include <hip/hip_runtime.h>
#include <hip/hip_bf16.h>

typedef __attribute__((ext_vector_type(2))) float v2f;
typedef __attribute__((ext_vector_type(8))) float v8f;

#define NUM_GRAPHS 64
#define DFEAT 128

// ---------------------------------------------------------------------------
// Degree counting (one thread per edge) + reciprocal clamp
// ---------------------------------------------------------------------------
__global__ void count_deg_kernel(const int* __restrict__ dst, float* __restrict__ deg, int E) {
    int e = blockIdx.x * blockDim.x + threadIdx.x;
    if (e < E) atomicAdd(deg + dst[e], 1.0f);
}

__global__ void inv_clamp_kernel(float* __restrict__ v, int n) {
    int i = blockIdx.x * blockDim.x + threadIdx.x;
    if (i < n) v[i] = 1.0f / fmaxf(v[i], 1.0f);
}

// ---------------------------------------------------------------------------
// Edge scatter-add: one wave (32 lanes) per edge, float4 per lane = 128 feats.
// Gather of x[src] is a coalesced 512B row read (L2-resident working set on
// MI455X's 192MB L2); scatter uses global_atomic_add_f32.
// ---------------------------------------------------------------------------
__global__ __launch_bounds__(256)
void scatter_add_kernel(const float* __restrict__ xsrc,
                        const int* __restrict__ src, const int* __restrict__ dst,
                        float* __restrict__ agg, int E) {
    int e = blockIdx.x * (blockDim.x >> 5) + (threadIdx.x >> 5);
    if (e >= E) return;
    int lane = threadIdx.x & 31;
    int s = src[e];
    int d = dst[e];
    const float4 v = ((const float4*)(xsrc + (size_t)s * DFEAT))[lane];
    float* p = agg + (size_t)d * DFEAT + lane * 4;
    atomicAdd(p + 0, v.x);
    atomicAdd(p + 1, v.y);
    atomicAdd(p + 2, v.z);
    atomicAdd(p + 3, v.w);
}

// ---------------------------------------------------------------------------
// WMMA fp32 GEMM: C[M,128] (+)= (rowscale .* A[M,128]) @ W[128,128] (+bias,relu)
// One wave per 16x16 output tile via V_WMMA_F32_16X16X4_F32 (32 K-steps).
// Block = 256 threads = 8 waves = all 8 N-tiles of one 16-row stripe.
// A fragment (16x4 f32, 2 VGPRs): lane half selects K-pair {k,k+1}/{k+2,k+3}.
// B fragment (4x16 f32, 2 VGPRs): same K-split, N = lane&15.
// C/D tile:  acc[v] -> row m0 + v + 8*half, col n0 + (lane&15).
// mode: 0 = init (C = bias), 1 = accumulate (C += existing [+bias])
// ---------------------------------------------------------------------------
__global__ __launch_bounds__(256)
void wmma_gemm128_kernel(const float* __restrict__ A, const float* __restrict__ rowscale,
                         const float* __restrict__ W, const float* __restrict__ bias,
                         float* __restrict__ C, int mode, int relu) {
    const int ntile = threadIdx.x >> 5;   // 0..7
    const int lane  = threadIdx.x & 31;
    const int r     = lane & 15;
    const int half  = lane >> 4;          // 0 or 1
    const int m0    = blockIdx.x << 4;
    const int n0    = ntile << 4;
    const int row   = m0 + r;

    const float* Arow = A + (size_t)row * DFEAT + 2 * half;
    const float  rs   = rowscale ? rowscale[row] : 1.0f;
    const float* Wp   = W + (size_t)(2 * half) * DFEAT + n0 + r;
    const float  bv   = bias ? bias[n0 + r] : 0.0f;

    v8f acc;
    if (mode == 0) {
        #pragma unroll
        for (int v = 0; v < 8; ++v) acc[v] = bv;
    } else {
        #pragma unroll
        for (int v = 0; v < 8; ++v)
            acc[v] = C[(size_t)(m0 + v + 8 * half) * DFEAT + n0 + r] + bv;
    }

    #pragma unroll 4
    for (int k = 0; k < DFEAT; k += 4) {
        v2f a;
        a.x = Arow[k]     * rs;
        a.y = Arow[k + 1] * rs;
        v2f b;
        b.x = Wp[(size_t)k * DFEAT];
        b.y = Wp[(size_t)(k + 1) * DFEAT];
        acc = __builtin_amdgcn_wmma_f32_16x16x4_f32(
            /*neg_a=*/false, a, /*neg_b=*/false, b,
            /*c_mod=*/(short)0, acc, /*reuse_a=*/false, /*reuse_b=*/false);
    }

    #pragma unroll
    for (int v = 0; v < 8; ++v) {
        float o = acc[v];
        if (relu) o = fmaxf(o, 0.0f);
        C[(size_t)(m0 + v + 8 * half) * DFEAT + n0 + r] = o;
    }
}

// ---------------------------------------------------------------------------
// Global mean-pool accumulation: one wave per node.
// ---------------------------------------------------------------------------
__global__ __launch_bounds__(256)
void pool_accum_kernel(const float* __restrict__ x, const int* __restrict__ batch,
                       float* __restrict__ pool, float* __restrict__ cnt,
                       int N, int coloff) {
    int n = blockIdx.x * (blockDim.x >> 5) + (threadIdx.x >> 5);
    if (n >= N) return;
    int lane = threadIdx.x & 31;
    int g = batch[n];
    const float4 v = ((const float4*)(x + (size_t)n * DFEAT))[lane];
    float* p = pool + (size_t)g * (2 * DFEAT) + coloff + lane * 4;
    atomicAdd(p + 0, v.x);
    atomicAdd(p + 1, v.y);
    atomicAdd(p + 2, v.z);
    atomicAdd(p + 3, v.w);
    if (lane == 0) atomicAdd(cnt + g, 1.0f);
}

__global__ void pool_finish_kernel(float* __restrict__ pool,
                                   const float* __restrict__ cu,
                                   const float* __restrict__ cd) {
    int i = blockIdx.x * blockDim.x + threadIdx.x;
    if (i >= NUM_GRAPHS * 2 * DFEAT) return;
    int g = i >> 8;       // /256
    int c = i & 255;
    float cnt = (c < DFEAT) ? cu[g] : cd[g];
    pool[i] /= fmaxf(cnt, 1.0f);
}

// ---------------------------------------------------------------------------
// Tiny head MLP (negligible FLOPs -> plain VALU)
// ---------------------------------------------------------------------------
__global__ void mlp1_kernel(const float* __restrict__ pool, const float* __restrict__ W1,
                            const float* __restrict__ b1, float* __restrict__ h) {
    int i = blockIdx.x * blockDim.x + threadIdx.x;
    if (i >= NUM_GRAPHS * DFEAT) return;
    int g = i >> 7;
    int j = i & 127;
    const float* p = pool + (size_t)g * (2 * DFEAT);
    float acc = b1[j];
    #pragma unroll 4
    for (int c = 0; c < 2 * DFEAT; ++c)
        acc = fmaf(p[c], W1[(size_t)c * DFEAT + j], acc);
    h[i] = fmaxf(acc, 0.0f);
}

__global__ void mlp2_logsoftmax_kernel(const float* __restrict__ h, const float* __restrict__ W2,
                                       const float* __restrict__ b2, float* __restrict__ out) {
    int g = blockIdx.x * blockDim.x + threadIdx.x;
    if (g >= NUM_GRAPHS) return;
    const float* hg = h + (size_t)g * DFEAT;
    float l0 = b2[0], l1 = b2[1];
    #pragma unroll 4
    for (int j = 0; j < DFEAT; ++j) {
        float v = hg[j];
        l0 = fmaf(v, W2[2 * j + 0], l0);
        l1 = fmaf(v, W2[2 * j + 1], l1);
    }
    float m = fmaxf(l0, l1);
    float lse = m + logf(expf(l0 - m) + expf(l1 - m));
    out[2 * g + 0] = l0 - lse;
    out[2 * g + 1] = l1 - lse;
}

// ---------------------------------------------------------------------------
// Host orchestration (graph-capture safe: only async stream ops)
// ---------------------------------------------------------------------------
extern "C" void kernel_launch(void* const* d_in, const int* in_sizes, int n_in,
                              void* d_out, int out_size, void* d_ws, size_t ws_size,
                              hipStream_t stream) {
    const float* x_user = (const float*)d_in[0];
    const float* x_drug = (const float*)d_in[1];
    const int*   ei_ud  = (const int*)d_in[2];
    const int*   ei_du  = (const int*)d_in[3];
    const int*   ei_uu  = (const int*)d_in[4];
    const int*   batch_user = (const int*)d_in[5];
    const int*   batch_drug = (const int*)d_in[6];

    const float* Wl1_ud = (const float*)d_in[7];
    const float* bl1_ud = (const float*)d_in[8];
    const float* Wr1_ud = (const float*)d_in[9];
    const float* Wl1_du = (const float*)d_in[10];
    const float* bl1_du = (const float*)d_in[11];
    const float* Wr1_du = (const float*)d_in[12];
    const float* Wl1_uu = (const float*)d_in[13];
    const float* bl1_uu = (const float*)d_in[14];
    const float* Wr1_uu = (const float*)d_in[15];
    const float* Wl2_ud = (const float*)d_in[16];
    const float* bl2_ud = (const float*)d_in[17];
    const float* Wr2_ud = (const float*)d_in[18];
    const float* Wl2_du = (const float*)d_in[19];
    const float* bl2_du = (const float*)d_in[20];
    const float* Wr2_du = (const float*)d_in[21];
    const float* Wl2_uu = (const float*)d_in[22];
    const float* bl2_uu = (const float*)d_in[23];
    const float* Wr2_uu = (const float*)d_in[24];
    const float* W1 = (const float*)d_in[25];
    const float* b1 = (const float*)d_in[26];
    const float* W2 = (const float*)d_in[27];
    const float* b2 = (const float*)d_in[28];

    const int N    = in_sizes[0] / DFEAT;      // 50000 (divisible by 16)
    const int E_ud = in_sizes[2] / 2;
    const int E_du = in_sizes[3] / 2;
    const int E_uu = in_sizes[4] / 2;
    const size_t NF = (size_t)N * DFEAT;

    // workspace layout
    float* ws = (float*)d_ws;
    float* agg    = ws;
    float* u1     = agg + NF;
    float* d1b    = u1 + NF;
    float* u2     = d1b + NF;
    float* d2b    = u2 + NF;
    float* inv_ud = d2b + NF;   // drug-side in-degree of ei_ud
    float* inv_du = inv_ud + N; // user-side in-degree of ei_du
    float* inv_uu = inv_du + N; // user-side in-degree of ei_uu
    float* pool   = inv_uu + N;                  // [64, 256]
    float* cnt_u  = pool + NUM_GRAPHS * 2 * DFEAT;
    float* cnt_d  = cnt_u + NUM_GRAPHS;
    float* hbuf   = cnt_d + NUM_GRAPHS;          // [64, 128]

    const int mtiles = N / 16;

    auto deg = [&](const int* ei, int E, float* buf) {
        hipMemsetAsync(buf, 0, (size_t)N * sizeof(float), stream);
        count_deg_kernel<<<(E + 255) / 256, 256, 0, stream>>>(ei + E, buf, E);
        inv_clamp_kernel<<<(N + 255) / 256, 256, 0, stream>>>(buf, N);
    };
    auto scatter = [&](const float* x, const int* ei, int E) {
        hipMemsetAsync(agg, 0, NF * sizeof(float), stream);
        scatter_add_kernel<<<(E + 7) / 8, 256, 0, stream>>>(x, ei, ei + E, agg, E);
    };
    auto gemm = [&](const float* A, const float* rs, const float* W, const float* b,
                    float* C, int mode, int relu) {
        wmma_gemm128_kernel<<<mtiles, 256, 0, stream>>>(A, rs, W, b, C, mode, relu);
    };

    // ---- in-degree reciprocals (mean folded into GEMM row scale) ----
    deg(ei_ud, E_ud, inv_ud);
    deg(ei_du, E_du, inv_du);
    deg(ei_uu, E_uu, inv_uu);

    // ---- Layer 1 ----
    // u1 = relu( sage(drug->user, du) + sage(user->user, uu) )
    scatter(x_drug, ei_du, E_du);
    gemm(agg,    inv_du, Wl1_du, bl1_du, u1, /*init*/0, 0);
    gemm(x_user, nullptr, Wr1_du, nullptr, u1, 1, 0);
    scatter(x_user, ei_uu, E_uu);
    gemm(agg,    inv_uu, Wl1_uu, bl1_uu, u1, 1, 0);
    gemm(x_user, nullptr, Wr1_uu, nullptr, u1, 1, /*relu*/1);
    // d1 = relu( sage(user->drug, ud) )
    scatter(x_user, ei_ud, E_ud);
    gemm(agg,    inv_ud, Wl1_ud, bl1_ud, d1b, 0, 0);
    gemm(x_drug, nullptr, Wr1_ud, nullptr, d1b, 1, /*relu*/1);

    // ---- Layer 2 (no relu) ----
    scatter(d1b, ei_du, E_du);
    gemm(agg, inv_du, Wl2_du, bl2_du, u2, 0, 0);
    gemm(u1,  nullptr, Wr2_du, nullptr, u2, 1, 0);
    scatter(u1, ei_uu, E_uu);
    gemm(agg, inv_uu, Wl2_uu, bl2_uu, u2, 1, 0);
    gemm(u1,  nullptr, Wr2_uu, nullptr, u2, 1, 0);
    scatter(u1, ei_ud, E_ud);
    gemm(agg, inv_ud, Wl2_ud, bl2_ud, d2b, 0, 0);
    gemm(d1b, nullptr, Wr2_ud, nullptr, d2b, 1, 0);

    // ---- Per-graph mean pool + head MLP + log_softmax ----
    hipMemsetAsync(pool, 0,
                   (NUM_GRAPHS * 2 * DFEAT + 2 * NUM_GRAPHS) * sizeof(float), stream);
    pool_accum_kernel<<<(N + 7) / 8, 256, 0, stream>>>(u2, batch_user, pool, cnt_u, N, 0);
    pool_accum_kernel<<<(N + 7) / 8, 256, 0, stream>>>(d2b, batch_drug, pool, cnt_d, N, DFEAT);
    pool_finish_kernel<<<(NUM_GRAPHS * 2 * DFEAT + 255) / 256, 256, 0, stream>>>(pool, cnt_u, cnt_d);
    mlp1_kernel<<<(NUM_GRAPHS * DFEAT + 255) / 256, 256, 0, stream>>>(pool, W1, b1, hbuf);
    mlp2_logsoftmax_kernel<<<1, NUM_GRAPHS, 0, stream>>>(hbuf, W2, b2, (float*)d_out);
}